// Model_63711544869124
// MI455X (gfx1250) — compile-verified
//
#include <hip/hip_runtime.h>
#include <cmath>

namespace {

constexpr int kImg   = 128;
constexpr int kFaces = 1024;
constexpr int kViews = 4;
constexpr float kNear = 0.1f;

// Per-view coefficient block: 6 pair-regions of kFaces float2 each (12K floats):
//   region 0: (a1,b1)   region 1: (c1,0)
//   region 2: (a2,b2)   region 3: (c2,0)
//   region 4: (ad,bd)   region 5: (cd,0)
constexpr int kCfView = 12 * kFaces;

typedef float v2f __attribute__((ext_vector_type(2)));
typedef float v8f __attribute__((ext_vector_type(8)));

struct ViewXform { float ex, ey, ez, pad; float R[9]; };
struct AllViews  { ViewXform v[kViews]; };

// ---------------------------------------------------------------------------
// Kernel 1: per (view, face) affine screen-space coefficients, stored as
// WMMA-A-fragment-ready (x,y) pairs. Degenerate faces get cd = -1e30 so the
// depth > NEAR test always fails (matches the reference's `ok` mask).
// ---------------------------------------------------------------------------
__global__ void face_coef_kernel(const float* __restrict__ verts,
                                 const int*   __restrict__ faces,
                                 float*       __restrict__ cf,
                                 AllViews vp, float width)
{
    int gid = blockIdx.x * blockDim.x + threadIdx.x;
    if (gid >= kViews * kFaces) return;
    int view = gid >> 10;
    int f    = gid & (kFaces - 1);
    const ViewXform& V = vp.v[view];

    float sx[3], sy[3], sz[3];
    #pragma unroll
    for (int k = 0; k < 3; ++k) {
        int vi = faces[f * 3 + k];
        float dx = verts[vi * 3 + 0] - V.ex;
        float dy = verts[vi * 3 + 1] - V.ey;
        float dz = verts[vi * 3 + 2] - V.ez;
        float vx = V.R[0] * dx + V.R[1] * dy + V.R[2] * dz;
        float vy = V.R[3] * dx + V.R[4] * dy + V.R[5] * dz;
        float vz = V.R[6] * dx + V.R[7] * dy + V.R[8] * dz;
        float zw = vz * width;
        sx[k] = vx / zw;
        sy[k] = vy / zw;
        sz[k] = vz;
    }
    float e1x = sx[1] - sx[0], e1y = sy[1] - sy[0];
    float e2x = sx[2] - sx[0], e2y = sy[2] - sy[0];
    float det = e1x * e2y - e1y * e2x;
    bool  ok  = fabsf(det) > 1e-8f;
    float inv = ok ? 1.0f / det : 0.0f;

    float a1 = e2y * inv, b1 = -e2x * inv;
    float c1 = -(a1 * sx[0] + b1 * sy[0]);
    float a2 = -e1y * inv, b2 = e1x * inv;
    float c2 = -(a2 * sx[0] + b2 * sy[0]);
    float dz1 = sz[1] - sz[0], dz2 = sz[2] - sz[0];
    float ad = a1 * dz1 + a2 * dz2;
    float bd = b1 * dz1 + b2 * dz2;
    float cd = ok ? (sz[0] + c1 * dz1 + c2 * dz2) : -1e30f;

    float* o = cf + view * kCfView;
    int fb = 2 * f;
    o[0 * kFaces + fb]     = a1;  o[0 * kFaces + fb + 1]  = b1;
    o[2 * kFaces + fb]     = c1;  o[2 * kFaces + fb + 1]  = 0.0f;
    o[4 * kFaces + fb]     = a2;  o[4 * kFaces + fb + 1]  = b2;
    o[6 * kFaces + fb]     = c2;  o[6 * kFaces + fb + 1]  = 0.0f;
    o[8 * kFaces + fb]     = ad;  o[8 * kFaces + fb + 1]  = bd;
    o[10 * kFaces + fb]    = cd;  o[10 * kFaces + fb + 1] = 0.0f;
}

// ---------------------------------------------------------------------------
// Kernel 2: WMMA rasterizer. Block = (view, image row): 8 waves x 16 pixels.
// Per 16-face x 16-pixel tile: 3x v_wmma_f32_16x16x4_f32 (w1, w2, depth).
// A-fragment (16x4 f32): lanes 0-15 hold K0,K1 ; lanes 16-31 hold K2,K3.
//   -> one unconditional ds_load_b64 per quantity: (a,b) or (c,0).
// B-fragment (4x16 f32): lanes 0-15 rows (px,py) ; lanes 16-31 rows (1,0).
// Z-reduce: u32 key = float bits of depth (positive floats order as uints);
// invisible -> 0xFFFFFFFF; strict < in increasing face order == argmin.
// ---------------------------------------------------------------------------
__global__ __launch_bounds__(256)
void raster_wmma_kernel(const float* __restrict__ cf,
                        const float* __restrict__ textures,
                        const float* __restrict__ refs,
                        float*       __restrict__ partials)
{
    __shared__ __align__(16) float lcf[kCfView];
    __shared__ float red[256];

    const int view = blockIdx.x >> 7;
    const int row  = blockIdx.x & (kImg - 1);
    const int tid  = threadIdx.x;

    // Stage this view's coefficient block into LDS (float4-coalesced, 48KB).
    {
        const float4* g = (const float4*)(cf + view * kCfView);
        float4*       s = (float4*)lcf;
        for (int i = tid; i < kCfView / 4; i += 256) s[i] = g[i];
    }
    __syncthreads();

    const int wave = tid >> 5;
    const int lane = tid & 31;
    const int nlo  = lane & 15;
    const int hi   = lane >> 4;
    const int col  = wave * 16 + nlo;

    const float s_col = ((col + 0.5f) / kImg) * 2.0f - 1.0f;
    const float s_row = ((row + 0.5f) / kImg) * 2.0f - 1.0f;
    const float px = s_col;
    const float py = -s_row;

    v2f bfrag;
    bfrag.x = hi ? 1.0f : px;   // rows K2=const term | K0 = px
    bfrag.y = hi ? 0.0f : py;   // rows K3=0          | K1 = py

    // Lane-constant pair-region bases: low lanes read (a,b), high lanes (c,0).
    const int base1 = (hi ? 2 : 0) * kFaces;
    const int base2 = (hi ? 6 : 4) * kFaces;
    const int base3 = (hi ? 10 : 8) * kFaces;

    unsigned bestKey = 0xFFFFFFFFu;
    int      bestF   = kFaces;

    #pragma unroll 2
    for (int t = 0; t < kFaces / 16; ++t) {
        const int fb = 2 * (t * 16 + nlo);
        v2f a1f = *(const v2f*)(lcf + base1 + fb);
        v2f a2f = *(const v2f*)(lcf + base2 + fb);
        v2f adf = *(const v2f*)(lcf + base3 + fb);

        v8f zero = {};
        v8f d1 = __builtin_amdgcn_wmma_f32_16x16x4_f32(false, a1f, false, bfrag,
                                                       (short)0, zero, false, false);
        v8f d2 = __builtin_amdgcn_wmma_f32_16x16x4_f32(false, a2f, false, bfrag,
                                                       (short)0, zero, false, false);
        v8f dd = __builtin_amdgcn_wmma_f32_16x16x4_f32(false, adf, false, bfrag,
                                                       (short)0, zero, false, false);
        #pragma unroll
        for (int r = 0; r < 8; ++r) {
            float w1  = d1[r];
            float w2  = d2[r];
            float dep = dd[r];
            float w0  = 1.0f - w1 - w2;
            bool vis = (w0 >= 0.0f) && (w1 >= 0.0f) && (w2 >= 0.0f) && (dep > kNear);
            unsigned key = vis ? __float_as_uint(dep) : 0xFFFFFFFFu;
            int fidx = t * 16 + hi * 8 + r;     // increasing face order
            bool take = key < bestKey;          // strict < keeps first hit
            bestF   = take ? fidx : bestF;
            bestKey = take ? key : bestKey;
        }
    }

    // Merge lane halves (same pixel, face sets {16t+0..7} vs {16t+8..15}).
    {
        unsigned oK = __shfl_xor(bestKey, 16, 32);
        int      oF = __shfl_xor(bestF,   16, 32);
        if (oK < bestKey || (oK == bestKey && oF < bestF)) {
            bestKey = oK; bestF = oF;
        }
    }

    // Texture sample + per-pixel loss (lanes 0-15 hold the final pixel state).
    float lossv = 0.0f;
    if (hi == 0) {
        float ref = refs[(view * kImg + row) * kImg + col];
        float c0 = 0.0f, c1 = 0.0f, c2 = 0.0f;
        bool hit = bestKey < 0x7F800000u;   // finite depth == reference's hit
        if (hit) {
            // Recover the winner's barycentrics from the LDS coefficients.
            int fb = 2 * bestF;
            float w1 = lcf[0 * kFaces + fb] * px + lcf[0 * kFaces + fb + 1] * py
                     + lcf[2 * kFaces + fb];
            float w2 = lcf[4 * kFaces + fb] * px + lcf[4 * kFaces + fb + 1] * py
                     + lcf[6 * kFaces + fb];
            float w0 = 1.0f - w1 - w2;

            float cx = fminf(fmaxf(w0, 0.0f), 1.0f) * 3.0f;
            float cy = fminf(fmaxf(w1, 0.0f), 1.0f) * 3.0f;
            float cz = fminf(fmaxf(w2, 0.0f), 1.0f) * 3.0f;
            int ix = (int)floorf(cx); ix = ix > 2 ? 2 : ix;
            int iy = (int)floorf(cy); iy = iy > 2 ? 2 : iy;
            int iz = (int)floorf(cz); iz = iz > 2 ? 2 : iz;
            float fx = cx - (float)ix;
            float fy = cy - (float)iy;
            float fz = cz - (float)iz;
            const float* tb = textures + (size_t)bestF * (4 * 4 * 4 * 3);
            #pragma unroll
            for (int dx = 0; dx < 2; ++dx)
            #pragma unroll
            for (int dy = 0; dy < 2; ++dy)
            #pragma unroll
            for (int dz = 0; dz < 2; ++dz) {
                float w = (dx ? fx : 1.0f - fx) * (dy ? fy : 1.0f - fy)
                        * (dz ? fz : 1.0f - fz);
                const float* tp = tb + (((ix + dx) * 4 + (iy + dy)) * 4 + (iz + dz)) * 3;
                c0 += w * tanhf(tp[0]);
                c1 += w * tanhf(tp[1]);
                c2 += w * tanhf(tp[2]);
            }
        }
        float e0 = c0 - ref, e1 = c1 - ref, e2 = c2 - ref;
        lossv = e0 * e0 + e1 * e1 + e2 * e2;
    }

    // Deterministic block tree reduction.
    red[tid] = lossv;
    __syncthreads();
    for (int s = 128; s > 0; s >>= 1) {
        if (tid < s) red[tid] += red[tid + s];
        __syncthreads();
    }
    if (tid == 0) partials[blockIdx.x] = red[0];
}

// ---------------------------------------------------------------------------
// Kernel 3: fixed-order final reduction of 512 block partials.
// ---------------------------------------------------------------------------
__global__ void reduce_kernel(const float* __restrict__ partials,
                              float* __restrict__ out)
{
    __shared__ float red[256];
    int tid = threadIdx.x;
    red[tid] = partials[tid] + partials[tid + 256];
    __syncthreads();
    for (int s = 128; s > 0; s >>= 1) {
        if (tid < s) red[tid] += red[tid + s];
        __syncthreads();
    }
    if (tid == 0) out[0] = red[0];
}

void make_view(double dist, double elev, double azim, ViewXform& V)
{
    const double d2r = M_PI / 180.0;
    double e = elev * d2r, a = azim * d2r;
    double ex = dist * cos(e) * sin(a);
    double ey = dist * sin(e);
    double ez = -dist * cos(e) * cos(a);
    double zx = -ex, zy = -ey, zz = -ez;
    double n = sqrt(zx * zx + zy * zy + zz * zz);
    zx /= n; zy /= n; zz /= n;
    // x = normalize(cross([0,1,0], z)) = (zz, 0, -zx) normalized
    double xx = zz, xy = 0.0, xz = -zx;
    n = sqrt(xx * xx + xz * xz);
    xx /= n; xz /= n;
    // y = normalize(cross(z, x))
    double yx = zy * xz - zz * xy;
    double yy = zz * xx - zx * xz;
    double yz = zx * xy - zy * xx;
    n = sqrt(yx * yx + yy * yy + yz * yz);
    yx /= n; yy /= n; yz /= n;
    V.ex = (float)ex; V.ey = (float)ey; V.ez = (float)ez; V.pad = 0.0f;
    V.R[0] = (float)xx; V.R[1] = (float)xy; V.R[2] = (float)xz;
    V.R[3] = (float)yx; V.R[4] = (float)yy; V.R[5] = (float)yz;
    V.R[6] = (float)zx; V.R[7] = (float)zy; V.R[8] = (float)zz;
}

} // namespace

extern "C" void kernel_launch(void* const* d_in, const int* in_sizes, int n_in,
                              void* d_out, int out_size, void* d_ws, size_t ws_size,
                              hipStream_t stream)
{
    (void)in_sizes; (void)n_in; (void)out_size; (void)ws_size;

    const float* verts    = (const float*)d_in[0];  // (1,1024,3) f32
    const int*   faces    = (const int*)  d_in[1];  // (1,1024,3) i32
    const float* textures = (const float*)d_in[2];  // (1,1024,4,4,4,3) f32
    const float* refs     = (const float*)d_in[3];  // (4,1,128,128) f32
    float*       out      = (float*)d_out;          // scalar loss

    float* cf       = (float*)d_ws;                 // 4 * 12K floats (192KB)
    float* partials = cf + kViews * kCfView;        // 512 floats

    AllViews vp;
    const double views[kViews][3] = {
        {2.83, 45.0, 0.0}, {2.0, 0.0, 90.0}, {3.46, 45.0, 45.0}, {3.0, 0.0, 0.0}
    };
    for (int i = 0; i < kViews; ++i)
        make_view(views[i][0], views[i][1], views[i][2], vp.v[i]);
    const float width = (float)tan(30.0 * M_PI / 180.0);

    face_coef_kernel<<<(kViews * kFaces + 255) / 256, 256, 0, stream>>>(
        verts, faces, cf, vp, width);
    raster_wmma_kernel<<<kImg * kViews, 256, 0, stream>>>(
        cf, textures, refs, partials);
    reduce_kernel<<<1, 256, 0, stream>>>(partials, out);
}